// GTM_90829968375862
// MI455X (gfx1250) — compile-verified
//
#include <hip/hip_runtime.h>
#include <math.h>

typedef float v2f __attribute__((ext_vector_type(2)));
typedef float v4f __attribute__((ext_vector_type(4)));
typedef float v8f __attribute__((ext_vector_type(8)));

#define D_DIM   512
#define K_REAL  360
#define K_PAD   368          // 23 x 16 WMMA row tiles
#define KT      23
#define NB      128          // samples per block
#define DC      16           // d-chunk staged in LDS
#define LDY     20           // padded LDS row stride (floats), 16B-aligned, conflict-free
#define LDT     20

// ---------------------------------------------------------------------------
// Kernel 1: Y = phi @ W  (phi computed analytically; sigma = 1/6 exactly for
// the 12x12 evenly spaced center grid), plus ynorm[k] = ||Y_k||^2.
// Pad rows (k >= 360): Y = 0, ynorm = 1e30 -> exp underflows to 0 in softmax.
// ---------------------------------------------------------------------------
__global__ void gtm_phiW(const float* __restrict__ W,
                         float* __restrict__ Yg,
                         float* __restrict__ ynorm) {
    __shared__ float phi[160];
    __shared__ float red[256];
    const int k   = blockIdx.x;
    const int tid = threadIdx.x;

    if (k >= K_REAL) {                       // uniform per block
        for (int d = tid; d < D_DIM; d += 256) Yg[k * D_DIM + d] = 0.0f;
        if (tid == 0) ynorm[k] = 1e30f;
        return;
    }

    if (tid < 145) {
        float p;
        if (tid < 144) {
            const int i0 = tid / 12, i1 = tid % 12;
            const float m0 = -11.0f / 12.0f + (float)i0 * (1.0f / 6.0f);
            const float m1 = -11.0f / 12.0f + (float)i1 * (1.0f / 6.0f);
            const int kx = k / 18, ky = k % 18;
            const float x0 = -1.0f + (float)kx * (2.0f / 19.0f);
            const float x1 = -1.0f + (float)ky * (2.0f / 17.0f);
            const float dx = x0 - m0, dy = x1 - m1;
            // phi = exp(-d2 / (2*sigma)), sigma = 1/6  ->  exp(-3*d2)
            p = expf(-3.0f * (dx * dx + dy * dy));
        } else {
            p = 1.0f;                         // bias column
        }
        phi[tid] = p;
    }
    __syncthreads();

    float sq = 0.0f;
    for (int d = tid; d < D_DIM; d += 256) {
        float acc = 0.0f;
        for (int j = 0; j < 145; ++j) acc = fmaf(phi[j], W[j * D_DIM + d], acc);
        Yg[k * D_DIM + d] = acc;
        sq += acc * acc;
    }
    red[tid] = sq;
    __syncthreads();
    for (int s = 128; s > 0; s >>= 1) {
        if (tid < s) red[tid] += red[tid + s];
        __syncthreads();
    }
    if (tid == 0) ynorm[k] = red[0];
}

// ---------------------------------------------------------------------------
// Kernel 2: fused  E = -(b/2)*max(||Y||^2 + ||t||^2 - 2*Y@t.T, 0),
// softmax over latent axis, R stores, and per-block sum(log s).
// Block = 128 samples x 368 latent rows, 8 waves, wave = 16 samples x 23 tiles.
// GEMM via V_WMMA_F32_16X16X4_F32, accumulators resident in VGPRs (23 x v8f).
// ---------------------------------------------------------------------------
__global__ void __launch_bounds__(256, 1)
gtm_main(const float* __restrict__ t,
         const float* __restrict__ Yg,
         const float* __restrict__ ynorm,
         const float* __restrict__ beta,
         float* __restrict__ out,          // out[0]=-ll (kernel 3), out+1 = R.T
         float* __restrict__ partial) {
    __shared__ float yl[K_PAD * LDY];      // 368*20 = 7360 floats
    __shared__ float tl[NB * LDT];         // 128*20 = 2560 floats
    __shared__ float ynL[K_PAD];
    __shared__ float tn[NB];
    __shared__ float wsum[8];

    const int tid  = threadIdx.x;
    const int n0   = blockIdx.x * NB;
    const int wave = tid >> 5;
    const int lane = tid & 31;
    const int half = lane >> 4;            // C-layout: M = 8*half + r
    const int l16  = lane & 15;            // C-layout: N = l16

    if (tid < NB) tn[tid] = 0.0f;
    for (int i = tid; i < K_PAD; i += 256) ynL[i] = ynorm[i];
    const float b = beta[0];
    __syncthreads();

    v8f zero = {};
    v8f acc[KT];
#pragma unroll
    for (int i = 0; i < KT; ++i) acc[i] = zero;

    for (int c = 0; c < D_DIM / DC; ++c) {
        const int dc = c * DC;
        // stage Y chunk: 368 rows x 16 floats (1472 float4 loads)
        for (int i = tid; i < K_PAD * (DC / 4); i += 256) {
            const int row = i >> 2, c4 = i & 3;
            v4f v = *(const v4f*)(Yg + row * D_DIM + dc + c4 * 4);
            *(v4f*)(yl + row * LDY + c4 * 4) = v;
        }
        // stage t chunk: 128 rows x 16 floats, fuse ||t||^2 accumulation
        for (int i = tid; i < NB * (DC / 4); i += 256) {
            const int row = i >> 2, c4 = i & 3;
            v4f v = *(const v4f*)(t + (size_t)(n0 + row) * D_DIM + dc + c4 * 4);
            *(v4f*)(tl + row * LDT + c4 * 4) = v;
            atomicAdd(&tn[row], v.x * v.x + v.y * v.y + v.z * v.z + v.w * v.w);
        }
        __syncthreads();

#pragma unroll
        for (int ds = 0; ds < DC; ds += 4) {
            // B (4x16): lane l16 = column n, halves hold k={0,1}/{2,3} pairs
            v2f bvec = *(const v2f*)(tl + (wave * 16 + l16) * LDT + ds + half * 2);
#pragma unroll
            for (int kt = 0; kt < KT; ++kt) {
                // A (16x4): lane l16 = row m, halves hold k={0,1}/{2,3} pairs
                v2f avec = *(const v2f*)(yl + (kt * 16 + l16) * LDY + ds + half * 2);
                acc[kt] = __builtin_amdgcn_wmma_f32_16x16x4_f32(
                    false, avec, false, bvec, (short)0, acc[kt], false, false);
            }
        }
        __syncthreads();
    }

    // ---- epilogue: E, softmax over latent axis (column split across lane^16)
    const int   ng  = n0 + wave * 16 + l16;
    const float tnv = tn[wave * 16 + l16];

    float colmax = -3.0e38f;
#pragma unroll
    for (int kt = 0; kt < KT; ++kt) {
#pragma unroll
        for (int r = 0; r < 8; ++r) {
            const int m = kt * 16 + half * 8 + r;
            const float d2 = fmaxf(ynL[m] + tnv - 2.0f * acc[kt][r], 0.0f);
            const float ev = -0.5f * b * d2;
            acc[kt][r] = ev;
            colmax = fmaxf(colmax, ev);
        }
    }
    colmax = fmaxf(colmax, __shfl_xor(colmax, 16, 32));

    float ssum = 0.0f;
#pragma unroll
    for (int kt = 0; kt < KT; ++kt) {
#pragma unroll
        for (int r = 0; r < 8; ++r) {
            const float p = __expf(acc[kt][r] - colmax);
            acc[kt][r] = p;
            ssum += p;
        }
    }
    ssum += __shfl_xor(ssum, 16, 32);
    const float inv = 1.0f / ssum;

    // R.T stores: out + 1 + n*360 + k ; 8 consecutive k per tile (b32 stores,
    // +1 float offset breaks 16B alignment; L2 (192MB) write-combines lines).
    float* outR = out + 1;
#pragma unroll
    for (int kt = 0; kt < KT; ++kt) {
        if (kt == KT - 1 && half) break;   // k = 360..367 are pad rows
        float* op = outR + (size_t)ng * K_REAL + kt * 16 + half * 8;
#pragma unroll
        for (int r = 0; r < 8; ++r) op[r] = acc[kt][r] * inv;
    }

    // per-block sum of log(s): deterministic fixed-order tree
    float v = (half == 0) ? logf(ssum) : 0.0f;
    v += __shfl_xor(v, 1, 32);
    v += __shfl_xor(v, 2, 32);
    v += __shfl_xor(v, 4, 32);
    v += __shfl_xor(v, 8, 32);
    if (lane == 0) wsum[wave] = v;
    __syncthreads();
    if (tid == 0) {
        float s8 = 0.0f;
        for (int i = 0; i < 8; ++i) s8 += wsum[i];
        partial[blockIdx.x] = s8;
    }
}

// ---------------------------------------------------------------------------
// Kernel 3: -ll = -( (D/2)*log(b/2pi) - log(K) + mean(log s) ), fixed-order sum.
// ---------------------------------------------------------------------------
__global__ void gtm_final(const float* __restrict__ partial,
                          const float* __restrict__ beta,
                          float* __restrict__ out, int nPart, int N) {
    __shared__ float red[256];
    const int tid = threadIdx.x;
    float s = 0.0f;
    for (int i = tid; i < nPart; i += 256) s += partial[i];
    red[tid] = s;
    __syncthreads();
    for (int k = 128; k > 0; k >>= 1) {
        if (tid < k) red[tid] += red[tid + k];
        __syncthreads();
    }
    if (tid == 0) {
        const float b = beta[0];
        const float C = 256.0f * logf(b / (2.0f * 3.14159265358979323846f))
                        - logf((float)K_REAL);
        out[0] = -(C + red[0] / (float)N);
    }
}

// ---------------------------------------------------------------------------
extern "C" void kernel_launch(void* const* d_in, const int* in_sizes, int n_in,
                              void* d_out, int out_size, void* d_ws, size_t ws_size,
                              hipStream_t stream) {
    const float* t    = (const float*)d_in[0];   // [N, 512]
    const float* W    = (const float*)d_in[1];   // [145, 512]
    const float* beta = (const float*)d_in[2];   // [1]
    float* out = (float*)d_out;                  // [1 + N*360]
    float* ws  = (float*)d_ws;

    const int N = in_sizes[0] / D_DIM;           // 131072

    float* partial = ws;                         // [N/NB] block partials
    float* Yg      = ws + 2048;                  // [368, 512]
    float* ynorm   = Yg + K_PAD * D_DIM;         // [368]

    gtm_phiW<<<K_PAD, 256, 0, stream>>>(W, Yg, ynorm);
    gtm_main<<<N / NB, 256, 0, stream>>>(t, Yg, ynorm, beta, out, partial);
    gtm_final<<<1, 256, 0, stream>>>(partial, beta, out, N / NB, N);
}